// LocalHolder2D_9826885173696
// MI455X (gfx1250) — compile-verified
//
#include <hip/hip_runtime.h>
#include <math.h>

#define TW 64
#define TH 64
#define HALO 3
#define LW (TW + 2 * HALO) /* 70 */
#define LH (TH + 2 * HALO) /* 70 */
#define NT 256
#define IMG_H 256
#define IMG_W 256
#define OUTS_PER_THREAD ((TW * TH) / NT) /* 16 */

// gfx1250 async global->LDS path (ASYNCcnt-tracked), with portable fallback.
#if defined(__HIP_DEVICE_COMPILE__) && defined(__gfx1250__) && \
    __has_builtin(__builtin_amdgcn_global_load_async_to_lds_b32)
#define USE_ASYNC_LDS 1
#else
#define USE_ASYNC_LDS 0
#endif

#if USE_ASYNC_LDS
typedef __attribute__((address_space(1))) int as1_int;
typedef __attribute__((address_space(3))) int as3_int;
#endif

__device__ __forceinline__ float max3f(float a, float b, float c) {
  return fmaxf(fmaxf(a, b), c);
}

__global__ __launch_bounds__(NT) void holder2d_kernel(
    const float* __restrict__ x, float* __restrict__ out,
    float w0, float w1, float w2) {
  __shared__ float A[LH * LW];
  __shared__ float B[LH * LW];

  const int tid = threadIdx.x;
  const int tx0 = blockIdx.x * TW;
  const int ty0 = blockIdx.y * TH;
  const int base = blockIdx.z * (IMG_H * IMG_W);
  const float NEG = -__builtin_huge_valf();

  // ---- Load (LH x LW) halo tile into LDS; out-of-image = -inf ----
  for (int idx = tid; idx < LH * LW; idx += NT) {
    const int i = idx / LW;
    const int j = idx - i * LW;
    const int gy = ty0 + i - HALO;
    const int gx = tx0 + j - HALO;
    const bool ok = ((unsigned)gy < (unsigned)IMG_H) && ((unsigned)gx < (unsigned)IMG_W);
#if USE_ASYNC_LDS
    if (ok) {
      __builtin_amdgcn_global_load_async_to_lds_b32(
          (as1_int*)(x + base + gy * IMG_W + gx), (as3_int*)(A + idx),
          /*offset=*/0, /*cpol=*/0);
    } else {
      A[idx] = NEG;
    }
#else
    A[idx] = ok ? x[base + gy * IMG_W + gx] : NEG;
#endif
  }
#if USE_ASYNC_LDS
#if __has_builtin(__builtin_amdgcn_s_wait_asynccnt)
  __builtin_amdgcn_s_wait_asynccnt(0);
#else
  asm volatile("s_wait_asynccnt 0" ::: "memory");
#endif
#endif
  __syncthreads();

  float acc[OUTS_PER_THREAD];
#pragma unroll
  for (int k = 0; k < OUTS_PER_THREAD; ++k) acc[k] = 0.0f;
  const float wgt[3] = {w0, w1, w2};

  // Stage s turns A = maxpool_{2s-1} region into A = maxpool_{2s+1} region:
  // maxpool3 applied repeatedly (separable hmax3 then vmax3), halo shrinking
  // by 1 each stage. Stage s valid region after vmax: [s, TH+5-s] x [s, TW+5-s].
#pragma unroll
  for (int s = 1; s <= 3; ++s) {
    {  // horizontal 3-max: rows [s-1, TH+6-s], cols [s, TW+5-s]
      const int r0 = s - 1;
      const int c0 = s;
      const int nr = (TH + 6 - s) - r0 + 1;
      const int nc = (TW + 5 - s) - c0 + 1;
      for (int idx = tid; idx < nr * nc; idx += NT) {
        const int i = r0 + idx / nc;
        const int j = c0 + idx % nc;
        const float* row = A + i * LW + j;
        B[i * LW + j] = max3f(row[-1], row[0], row[1]);
      }
    }
    __syncthreads();
    {  // vertical 3-max: rows [s, TH+5-s], cols [s, TW+5-s]
      const int r0 = s;
      const int c0 = s;
      const int nr = (TH + 5 - s) - r0 + 1;
      const int nc = (TW + 5 - s) - c0 + 1;
      for (int idx = tid; idx < nr * nc; idx += NT) {
        const int i = r0 + idx / nc;
        const int j = c0 + idx % nc;
        const float* col = B + i * LW + j;
        A[i * LW + j] = max3f(col[-LW], col[0], col[LW]);
      }
    }
    __syncthreads();
    // Accumulate w[s-1] * log10(maxpool_{2s+1}) at the tile centers.
    // (Next stage's hmax only reads A / writes B, so no extra barrier needed.)
    const float wv = wgt[s - 1];
#pragma unroll
    for (int k = 0; k < OUTS_PER_THREAD; ++k) {
      const int o = tid + k * NT;
      const int oy = o >> 6;        // TW == 64
      const int ox = o & (TW - 1);
      acc[k] += wv * log10f(A[(oy + HALO) * LW + (ox + HALO)]);
    }
  }

#pragma unroll
  for (int k = 0; k < OUTS_PER_THREAD; ++k) {
    const int o = tid + k * NT;
    const int oy = o >> 6;
    const int ox = o & (TW - 1);
    out[base + (ty0 + oy) * IMG_W + (tx0 + ox)] = acc[k];
  }
}

extern "C" void kernel_launch(void* const* d_in, const int* in_sizes, int n_in,
                              void* d_out, int out_size, void* d_ws, size_t ws_size,
                              hipStream_t stream) {
  const float* x = (const float*)d_in[0];
  float* out = (float*)d_out;
  const int BC = in_sizes[0] / (IMG_H * IMG_W);  // 8*64 = 512

  // Closed-form OLS slope weights: w_i = (xs_i - mean) / sum((xs_j - mean)^2),
  // xs_i = log10(k_i / (H*W)), k = 3,5,7.
  double xs[3];
  double m = 0.0;
  for (int i = 0; i < 3; ++i) {
    const double k = 2.0 * (i + 1) + 1.0;
    xs[i] = log10(k / (double)(IMG_H * IMG_W));
    m += xs[i];
  }
  m /= 3.0;
  double ss = 0.0;
  for (int i = 0; i < 3; ++i) {
    xs[i] -= m;
    ss += xs[i] * xs[i];
  }
  const float w0 = (float)(xs[0] / ss);
  const float w1 = (float)(xs[1] / ss);
  const float w2 = (float)(xs[2] / ss);

  dim3 grid(IMG_W / TW, IMG_H / TH, BC);  // 4 x 4 x 512
  holder2d_kernel<<<grid, NT, 0, stream>>>(x, out, w0, w1, w2);
}